// GCNLayer_46162308497632
// MI455X (gfx1250) — compile-verified
//
#include <hip/hip_runtime.h>
#include <hip/hip_bf16.h>
#include <cstdint>

#define D 128
#define BN_EPS 1e-5f

typedef float v2f __attribute__((ext_vector_type(2)));
typedef float v8f __attribute__((ext_vector_type(8)));

// ---------------------------------------------------------------------------
// K0: zero a float range (agg / deg / column-stat accumulators)
// ---------------------------------------------------------------------------
__global__ void zero_kernel(float* __restrict__ p, long long n) {
    long long i = (long long)blockIdx.x * blockDim.x + threadIdx.x;
    if (i < n) p[i] = 0.0f;
}

// ---------------------------------------------------------------------------
// K1: edge scatter. One wave per edge: 32 lanes x float4 = 128 columns.
// agg[dst] += feature[src]; deg[dst] += 1.  All traffic is L2-resident
// (node table = 25.6MB << 192MB L2); unsafeAtomicAdd -> global_atomic_add_f32.
// ---------------------------------------------------------------------------
__global__ void edge_scatter_kernel(const long long* __restrict__ src,
                                    const long long* __restrict__ dst,
                                    const float* __restrict__ feat,
                                    float* __restrict__ agg,
                                    float* __restrict__ deg,
                                    int n_edges) {
    const int e = blockIdx.x * (blockDim.x >> 5) + (threadIdx.x >> 5);
    if (e >= n_edges) return;
    const int lane = threadIdx.x & 31;
    const long long s = src[e];
    const long long d = dst[e];
    const float4 v = *reinterpret_cast<const float4*>(feat + s * D + lane * 4);
    float* ap = agg + d * D + lane * 4;
    unsafeAtomicAdd(ap + 0, v.x);
    unsafeAtomicAdd(ap + 1, v.y);
    unsafeAtomicAdd(ap + 2, v.z);
    unsafeAtomicAdd(ap + 3, v.w);
    if (lane == 0) unsafeAtomicAdd(deg + d, 1.0f);
}

// ---------------------------------------------------------------------------
// K2: lin = (agg / deg) @ W^T + b using V_WMMA_F32_16X16X4_F32.
// One wave computes a 16x128 output stripe (8 accumulator tiles of v8f).
// W (64KB) is staged into LDS once per workgroup; degree normalization is
// fused into the A-fragment load; bias fused into the store.
//
// Fragment layouts (ISA 7.12.2, 32-bit):
//   A 16x4 : lane%16 = M, VGPRs = K{0,1} (lanes 0-15) / K{2,3} (lanes 16-31)
//   B 4x16 : lane%16 = N, VGPRs = K{0,1} (lanes 0-15) / K{2,3} (lanes 16-31)
//   C 16x16: lane%16 = N, VGPR r = row r (lanes 0-15) / row 8+r (lanes 16-31)
// ---------------------------------------------------------------------------
__global__ void __launch_bounds__(256)
gemm_wmma_kernel(const float* __restrict__ agg, const float* __restrict__ deg,
                 const float* __restrict__ W, const float* __restrict__ bias,
                 float* __restrict__ lin, int n_nodes) {
    __shared__ float sW[D * D];  // 64 KB: W row-major [out][in]
    for (int i = threadIdx.x * 4; i < D * D; i += blockDim.x * 4) {
        *reinterpret_cast<float4*>(sW + i) = *reinterpret_cast<const float4*>(W + i);
    }
    __syncthreads();

    const int n_tiles = (n_nodes + 15) / 16;
    const int mt = blockIdx.x * (blockDim.x >> 5) + (threadIdx.x >> 5);
    if (mt >= n_tiles) return;  // wave-uniform: EXEC stays all-ones for WMMA

    const int lane = threadIdx.x & 31;
    const int l16  = lane & 15;
    const int hi   = lane >> 4;  // 0: K{0,1}/rows 0-7, 1: K{2,3}/rows 8-15

    const int   rowA = mt * 16 + l16;           // row this lane supplies to A
    const int   rowL = min(rowA, n_nodes - 1);  // clamp for tail tile
    const float dv   = deg[rowL];
    const float rdeg = (rowA < n_nodes && dv > 0.0f) ? (1.0f / dv) : 0.0f;
    const float* __restrict__ arow = agg + (size_t)rowL * D + 2 * hi;

    v8f acc[8] = {};  // 8 N-tiles x 8 VGPRs = full 16x128 stripe

    for (int kk = 0; kk < D; kk += 4) {
        const float2 av = *reinterpret_cast<const float2*>(arow + kk);
        v2f a;
        a.x = av.x * rdeg;  // fused mean-aggregation
        a.y = av.y * rdeg;
#pragma unroll
        for (int nt = 0; nt < 8; ++nt) {
            // B[k][n] = W[n][k]; lane holds column n = nt*16+l16, K = kk+2*hi+{0,1}
            const float2 bv =
                *reinterpret_cast<const float2*>(sW + (nt * 16 + l16) * D + kk + 2 * hi);
            v2f b;
            b.x = bv.x;
            b.y = bv.y;
            acc[nt] = __builtin_amdgcn_wmma_f32_16x16x4_f32(
                /*neg_a=*/false, a, /*neg_b=*/false, b,
                /*c_mod=*/(short)0, acc[nt], /*reuse_a=*/false, /*reuse_b=*/false);
        }
    }

#pragma unroll
    for (int nt = 0; nt < 8; ++nt) {
        const int   j  = nt * 16 + l16;
        const float bj = bias[j];
#pragma unroll
        for (int r = 0; r < 8; ++r) {
            const int row = mt * 16 + hi * 8 + r;
            if (row < n_nodes) lin[(size_t)row * D + j] = acc[nt][r] + bj;
        }
    }
}

// ---------------------------------------------------------------------------
// K3: per-column sum / sum-of-squares over the node dimension.
// Coalesced: threads 0-127 cover one row, 128-255 the next.
// ---------------------------------------------------------------------------
__global__ void col_stats_kernel(const float* __restrict__ lin,
                                 float* __restrict__ colsum,
                                 float* __restrict__ colsumsq, int n_nodes) {
    const int col  = threadIdx.x & (D - 1);
    const int half = threadIdx.x >> 7;
    const int rows_per = (n_nodes + gridDim.x - 1) / gridDim.x;
    const int r0 = blockIdx.x * rows_per;
    const int r1 = min(r0 + rows_per, n_nodes);
    float s = 0.0f, s2 = 0.0f;
    for (int r = r0 + half; r < r1; r += 2) {
        const float v = lin[(size_t)r * D + col];
        s += v;
        s2 += v * v;
    }
    __shared__ float sh[256], sh2[256];
    sh[threadIdx.x]  = s;
    sh2[threadIdx.x] = s2;
    __syncthreads();
    if (threadIdx.x < 128) {
        unsafeAtomicAdd(colsum + col, sh[threadIdx.x] + sh[threadIdx.x + 128]);
        unsafeAtomicAdd(colsumsq + col, sh2[threadIdx.x] + sh2[threadIdx.x + 128]);
    }
}

// ---------------------------------------------------------------------------
// K4: fold BN stats into per-column scale/shift.
// ---------------------------------------------------------------------------
__global__ void finalize_stats_kernel(const float* __restrict__ colsum,
                                      const float* __restrict__ colsumsq,
                                      const float* __restrict__ gamma,
                                      const float* __restrict__ beta,
                                      float* __restrict__ scale,
                                      float* __restrict__ shift, int n_nodes) {
    const int   j     = threadIdx.x;
    const float inv_n = 1.0f / (float)n_nodes;
    const float mu    = colsum[j] * inv_n;
    float       var   = colsumsq[j] * inv_n - mu * mu;  // biased, matches reference
    var = fmaxf(var, 0.0f);
    const float sc = gamma[j] / sqrtf(var + BN_EPS);
    scale[j] = sc;
    shift[j] = beta[j] - mu * sc;
}

// ---------------------------------------------------------------------------
// K5: out = feature + relu(lin * scale + shift), float4-vectorized.
// ---------------------------------------------------------------------------
__global__ void fused_out_kernel(const float* __restrict__ feat,
                                 const float* __restrict__ lin,
                                 const float* __restrict__ scale,
                                 const float* __restrict__ shift,
                                 float* __restrict__ out, long long n4) {
    const long long i = (long long)blockIdx.x * blockDim.x + threadIdx.x;
    if (i >= n4) return;
    const int g = (int)(i & (D / 4 - 1));  // float4 group within the row
    const float4 f  = reinterpret_cast<const float4*>(feat)[i];
    const float4 l  = reinterpret_cast<const float4*>(lin)[i];
    const float4 sc = reinterpret_cast<const float4*>(scale)[g];
    const float4 sh = reinterpret_cast<const float4*>(shift)[g];
    float4 o;
    o.x = f.x + fmaxf(l.x * sc.x + sh.x, 0.0f);
    o.y = f.y + fmaxf(l.y * sc.y + sh.y, 0.0f);
    o.z = f.z + fmaxf(l.z * sc.z + sh.z, 0.0f);
    o.w = f.w + fmaxf(l.w * sc.w + sh.w, 0.0f);
    reinterpret_cast<float4*>(out)[i] = o;
}

// ---------------------------------------------------------------------------
extern "C" void kernel_launch(void* const* d_in, const int* in_sizes, int n_in,
                              void* d_out, int out_size, void* d_ws, size_t ws_size,
                              hipStream_t stream) {
    const float*      feature = (const float*)d_in[0];
    const float*      W       = (const float*)d_in[1];
    const float*      b       = (const float*)d_in[2];
    const float*      gamma   = (const float*)d_in[3];
    const float*      beta    = (const float*)d_in[4];
    const long long*  src     = (const long long*)d_in[5];  // int64 per reference
    const long long*  dst     = (const long long*)d_in[6];
    float*            out     = (float*)d_out;

    const int n_nodes = in_sizes[0] / D;
    const int n_edges = in_sizes[5];

    // workspace layout (floats)
    float* ws       = (float*)d_ws;
    float* agg      = ws;                              // N*D
    float* lin      = agg + (size_t)n_nodes * D;       // N*D
    float* deg      = lin + (size_t)n_nodes * D;       // N
    float* colsum   = deg + n_nodes;                   // D
    float* colsumsq = colsum + D;                      // D
    float* scale    = colsumsq + D;                    // D
    float* shift    = scale + D;                       // D

    // K0: zero accumulators
    const long long n0 = (long long)n_nodes * D;
    zero_kernel<<<(unsigned)((n0 + 255) / 256), 256, 0, stream>>>(agg, n0);
    const long long n1 = (long long)n_nodes + 2 * D;
    zero_kernel<<<(unsigned)((n1 + 255) / 256), 256, 0, stream>>>(deg, n1);

    // K1: edge scatter (one wave per edge, 8 edges per 256-thread block)
    edge_scatter_kernel<<<(n_edges + 7) / 8, 256, 0, stream>>>(src, dst, feature,
                                                               agg, deg, n_edges);

    // K2: WMMA GEMM (one wave per 16-row tile, 8 tiles per block)
    const int n_tiles = (n_nodes + 15) / 16;
    gemm_wmma_kernel<<<(n_tiles + 7) / 8, 256, 0, stream>>>(agg, deg, W, b, lin,
                                                            n_nodes);

    // K3/K4: BatchNorm statistics
    col_stats_kernel<<<128, 256, 0, stream>>>(lin, colsum, colsumsq, n_nodes);
    finalize_stats_kernel<<<1, D, 0, stream>>>(colsum, colsumsq, gamma, beta,
                                               scale, shift, n_nodes);

    // K5: fused BN-apply + relu + residual
    const long long n4 = (long long)n_nodes * D / 4;
    fused_out_kernel<<<(unsigned)((n4 + 255) / 256), 256, 0, stream>>>(
        feature, lin, scale, shift, out, n4);
}